// Decoder_3418793967936
// MI455X (gfx1250) — compile-verified
//
#include <hip/hip_runtime.h>
#include <stdint.h>

typedef __attribute__((ext_vector_type(2))) float v2f;
typedef __attribute__((ext_vector_type(8))) float v8f;

#define HH 512
#define EE 512
#define SEQ 512
#define BB 64
#define VOCAB 32000

// ---------------- kernel 0: embedding gather + h.Wh reduction ----------------
__global__ void k_embed(const int* __restrict__ tokens,
                        const float* __restrict__ emb_table,
                        const float* __restrict__ hidden,
                        const float* __restrict__ energy_W,
                        float* __restrict__ xbuf,    // (64,1024) = [emb | ctx]
                        float* __restrict__ outbuf,  // (64,1536) = [emb | h_new | ctx]
                        float* __restrict__ hwh)     // (64)
{
    const int b = blockIdx.x, t = threadIdx.x;
    __shared__ float red[512];
    float emb = emb_table[(size_t)tokens[b] * EE + t];
    xbuf[b * 1024 + t] = emb;
    outbuf[b * 1536 + t] = emb;
    float p = hidden[b * HH + t] * energy_W[t];   // Wh = energy_W[0,:H]
    red[t] = p;
    __syncthreads();
    for (int off = 256; off > 0; off >>= 1) {
        if (t < off) red[t] += red[t + off];
        __syncthreads();
    }
    if (t == 0) hwh[b] = red[0];
}

// ---------------- kernel 1: attention energies, one wave per (s,b) ----------------
__global__ void k_energy(const float* __restrict__ enc,
                         const float* __restrict__ energy_W,
                         const float* __restrict__ energy_b,
                         const float* __restrict__ hwh,
                         float* __restrict__ e)    // (S,B)
{
    const int wid  = (blockIdx.x * blockDim.x + threadIdx.x) >> 5;
    const int lane = threadIdx.x & 31;
    const int s = wid >> 6, b = wid & 63;
    const float* row = enc + ((size_t)s * BB + b) * HH;
    const float* We  = energy_W + HH;             // We = energy_W[0,H:]
    float part = 0.f;
#pragma unroll
    for (int j = 0; j < 16; ++j) {
        int idx = lane + 32 * j;
        part = fmaf(row[idx], We[idx], part);
    }
#pragma unroll
    for (int m = 16; m > 0; m >>= 1) part += __shfl_xor(part, m, 32);
    if (lane == 0)
        e[s * BB + b] = fmaxf(part + hwh[b] + energy_b[0], 0.f);
}

// ---------------- kernel 2: softmax over S + context, one block per b ----------------
__global__ void k_softmax_ctx(const float* __restrict__ enc,
                              const float* __restrict__ e,
                              float* __restrict__ xbuf,
                              float* __restrict__ outbuf)
{
    const int b = blockIdx.x, t = threadIdx.x;
    __shared__ float red[512];
    __shared__ float attnb[512];
    float ev = e[t * BB + b];
    red[t] = ev; __syncthreads();
    for (int off = 256; off > 0; off >>= 1) {
        if (t < off) red[t] = fmaxf(red[t], red[t + off]);
        __syncthreads();
    }
    float m = red[0]; __syncthreads();
    float p = __expf(ev - m);
    red[t] = p; __syncthreads();
    for (int off = 256; off > 0; off >>= 1) {
        if (t < off) red[t] += red[t + off];
        __syncthreads();
    }
    float l = red[0]; __syncthreads();
    attnb[t] = p / l;
    __syncthreads();
    float ctx = 0.f;
    for (int s = 0; s < SEQ; ++s)
        ctx = fmaf(attnb[s], enc[((size_t)s * BB + b) * HH + t], ctx);
    xbuf[b * 1024 + HH + t] = ctx;
    outbuf[b * 1536 + 1024 + t] = ctx;
}

// ---------------- WMMA f32 GEMM: D(64,N) = A(64,K) * W(N,K)^T + bias ----------------
// 8 waves/block; wave owns 16 N-columns x all 64 M-rows (4 wmma accumulators).
// BOTH the W tile (128 x 32 f32) and the A chunk (64 x 32 f32) are staged into LDS
// via async global->LDS DMA, double-buffered with s_wait_asynccnt + split barriers.
// Steady-state inner loop: ds_load_b64 (B frag) + 4x v_wmma, A frags preloaded in
// registers from LDS (no global loads / loadcnt stalls inside the K loop).
__global__ __launch_bounds__(256) void k_gemm_wt(
    const float* __restrict__ A,     // (64,K) row-major
    const float* __restrict__ W,     // (N,K) row-major
    const float* __restrict__ bias,  // (N)
    float* __restrict__ D,           // (64,N) row-major
    int N, int K)
{
    __shared__ float wtile[2][128 * 36];   // 36-float pitch: bank-conflict free
    __shared__ float atile[2][64 * 36];
    const int tid  = threadIdx.x;
    const int wave = tid >> 5;
    const int lane = tid & 31;
    const int ln   = lane & 15;            // N (and M) index within 16-tile
    const int koff = (lane >> 4) * 2;      // half-wave K offset {0,2}
    const int n0   = blockIdx.x * 128;
    const int nchunks = K >> 5;

    auto load_chunk = [&](int c, int buf) {
        const int k0 = c << 5;
        // W tile: 128 rows x 128B  -> 1024 x 16B slots, 4 per thread
#pragma unroll
        for (int i = 0; i < 4; ++i) {
            int slot = tid + i * 256;
            int row  = slot >> 3;          // 0..127
            int c4   = slot & 7;           // 16B unit within 128B row segment
            const float* g = W + (size_t)(n0 + row) * K + k0 + c4 * 4;
            float* lp = &wtile[buf][row * 36 + c4 * 4];
            unsigned loff = (unsigned)(uintptr_t)lp;   // low 32 bits = LDS offset
            asm volatile("global_load_async_to_lds_b128 %0, %1, off"
                         :: "v"(loff), "v"(g) : "memory");
        }
        // A chunk: 64 rows x 128B -> 512 x 16B slots, 2 per thread
#pragma unroll
        for (int i = 0; i < 2; ++i) {
            int slot = tid + i * 256;
            int row  = slot >> 3;          // 0..63
            int c4   = slot & 7;
            const float* g = A + (size_t)row * K + k0 + c4 * 4;
            float* lp = &atile[buf][row * 36 + c4 * 4];
            unsigned loff = (unsigned)(uintptr_t)lp;
            asm volatile("global_load_async_to_lds_b128 %0, %1, off"
                         :: "v"(loff), "v"(g) : "memory");
        }
    };

    v8f acc[4];
#pragma unroll
    for (int mt = 0; mt < 4; ++mt)
#pragma unroll
        for (int r = 0; r < 8; ++r) acc[mt][r] = 0.f;

    load_chunk(0, 0);
    for (int c = 0; c < nchunks; ++c) {
        const int buf = c & 1;
        if (c + 1 < nchunks) {
            load_chunk(c + 1, buf ^ 1);
            asm volatile("s_wait_asynccnt 6" ::: "memory");  // chunk c's 6 DMAs done
        } else {
            asm volatile("s_wait_asynccnt 0" ::: "memory");
        }
        __syncthreads();

        // Preload all A fragments of this chunk from LDS into registers.
        v2f af[4][8];
#pragma unroll
        for (int mt = 0; mt < 4; ++mt)
#pragma unroll
            for (int ks = 0; ks < 8; ++ks)
                af[mt][ks] = *(const v2f*)(&atile[buf][(mt * 16 + ln) * 36 + ks * 4 + koff]);

        const float* wl = &wtile[buf][(wave * 16 + ln) * 36];
#pragma unroll
        for (int ks = 0; ks < 8; ++ks) {
            // B frag: lanes0-15 rows K={0,1}, lanes16-31 rows K={2,3}, col = ln
            v2f bf = *(const v2f*)(wl + ks * 4 + koff);
#pragma unroll
            for (int mt = 0; mt < 4; ++mt)
                acc[mt] = __builtin_amdgcn_wmma_f32_16x16x4_f32(
                    false, af[mt][ks], false, bf, (short)0, acc[mt], false, false);
        }
        __syncthreads();
    }

    const int n = n0 + wave * 16 + ln;
    const float bv = bias ? bias[n] : 0.f;
    const int mbase = (lane >> 4) * 8;     // VGPR r -> M=r (lanes 0-15) / r+8 (16-31)
#pragma unroll
    for (int mt = 0; mt < 4; ++mt)
#pragma unroll
        for (int r = 0; r < 8; ++r)
            D[(size_t)(mt * 16 + mbase + r) * N + n] = acc[mt][r] + bv;
}

// ---------------- kernel 5: GRU elementwise ----------------
__global__ void k_gru(const float* __restrict__ gi, const float* __restrict__ gh,
                      const float* __restrict__ hidden,
                      float* __restrict__ outbuf, float* __restrict__ hnew_out)
{
    const int b = blockIdx.x, j = threadIdx.x;
    const float* gib = gi + b * 1536;
    const float* ghb = gh + b * 1536;
    float r = 1.f / (1.f + __expf(-(gib[j] + ghb[j])));
    float z = 1.f / (1.f + __expf(-(gib[512 + j] + ghb[512 + j])));
    float n = tanhf(gib[1024 + j] + r * ghb[1024 + j]);
    float hb = hidden[b * HH + j];
    float hn = (1.f - z) * n + z * hb;
    outbuf[b * 1536 + 512 + j] = hn;     // middle section of [emb|h_new|ctx]
    hnew_out[b * HH + j] = hn;
}

extern "C" void kernel_launch(void* const* d_in, const int* in_sizes, int n_in,
                              void* d_out, int out_size, void* d_ws, size_t ws_size,
                              hipStream_t stream)
{
    const int*   tokens   = (const int*)d_in[0];
    const float* hidden   = (const float*)d_in[1];
    const float* enc      = (const float*)d_in[2];
    const float* emb      = (const float*)d_in[3];
    const float* energy_W = (const float*)d_in[4];
    const float* energy_b = (const float*)d_in[5];
    const float* W_ih     = (const float*)d_in[6];
    const float* W_hh     = (const float*)d_in[7];
    const float* b_ih     = (const float*)d_in[8];
    const float* b_hh     = (const float*)d_in[9];
    const float* fc_W     = (const float*)d_in[10];
    const float* fc_b     = (const float*)d_in[11];
    float* out = (float*)d_out;

    float* ws   = (float*)d_ws;
    float* e    = ws;             // 32768 floats (S,B)
    float* hwh  = ws + 32768;     // 64 (+pad)
    float* xbuf = ws + 33024;     // 65536  (64,1024)
    float* outb = ws + 98560;     // 98304  (64,1536)
    float* gi   = ws + 196864;    // 98304  (64,1536)
    float* gh   = ws + 295168;    // 98304  (64,1536)

    k_embed<<<BB, 512, 0, stream>>>(tokens, emb, hidden, energy_W, xbuf, outb, hwh);
    k_energy<<<(SEQ * BB) / 8, 256, 0, stream>>>(enc, energy_W, energy_b, hwh, e);
    k_softmax_ctx<<<BB, 512, 0, stream>>>(enc, e, xbuf, outb);
    k_gemm_wt<<<1536 / 128, 256, 0, stream>>>(xbuf, W_ih, b_ih, gi, 1536, 1024);
    k_gemm_wt<<<1536 / 128, 256, 0, stream>>>(hidden, W_hh, b_hh, gh, 1536, 512);
    k_gru<<<BB, 512, 0, stream>>>(gi, gh, hidden, outb, out + (size_t)BB * VOCAB);
    k_gemm_wt<<<VOCAB / 128, 256, 0, stream>>>(outb, fc_W, fc_b, out, VOCAB, 1536);
}